// DifferentiableRobotModel_57080115364384
// MI455X (gfx1250) — compile-verified
//
#include <hip/hip_runtime.h>

#define NLINK 32
#define TPB   256

typedef float v4f __attribute__((ext_vector_type(4)));

__device__ __forceinline__ float sgnf(float v) {
    // matches jnp.where(v >= 0, 1, -1)
    return v >= 0.0f ? 1.0f : -1.0f;
}

__global__ __launch_bounds__(TPB) void fk_chain_kernel(
    const float* __restrict__ q,          // [B, 32]
    const float* __restrict__ link_trans, // [32, 3]
    const float* __restrict__ link_rot,   // [32, 3, 3]
    const float* __restrict__ joint_axes, // [32, 3]
    float* __restrict__ out,              // [32, B, 7]
    int nB)
{
    // LDS layout (floats): [0,288) Rf, [288,384) tf, [384,480) axes -> 1920 B
    __shared__ __align__(16) float smem[480];

    const int t = threadIdx.x;
    const int b = blockIdx.x * TPB + t;

    // Kick off this thread's q-row prefetch (one 128B line per thread) early so
    // it overlaps the async staging + barrier below.
    const v4f* __restrict__ q4 =
        reinterpret_cast<const v4f*>(q + (size_t)b * NLINK);
    __builtin_prefetch(q4, 0, 0);   // -> global_prefetch_b8

    // ---- CDNA5 async global->LDS staging of the wave-uniform link constants.
    // 120 x 16B chunks: 72 for link_rot, 24 for link_trans, 24 for joint_axes.
    if (t < 120) {
        // Branchless source select.
        const float* g = (t < 72) ? (link_rot + t * 4)
                       : (t < 96) ? (link_trans + (t - 72) * 4)
                                  : (joint_axes + (t - 96) * 4);
        unsigned ldsaddr = (unsigned)(unsigned long long)(smem + t * 4);
        asm volatile("global_load_async_to_lds_b128 %0, %1, off"
                     :: "v"(ldsaddr), "v"(g) : "memory");
    }
    asm volatile("s_wait_asynccnt 0x0" ::: "memory");
    __syncthreads();

    if (b >= nB) return;

    // Carried pose
    float R[9] = {1.f, 0.f, 0.f, 0.f, 1.f, 0.f, 0.f, 0.f, 1.f};
    float tp[3] = {0.f, 0.f, 0.f};

    const size_t lstep = (size_t)nB * 7;   // link-to-link output stride (floats)
    float* o = out + (size_t)b * 7;        // advanced by lstep per link

#pragma unroll 1
    for (int c = 0; c < 8; ++c) {
        v4f qv = q4[c];
        float qarr[4] = {qv.x, qv.y, qv.z, qv.w};
#pragma unroll
        for (int j = 0; j < 4; ++j) {
            const int i = c * 4 + j;
            const float qi = qarr[j];

            // ---- wave-uniform constants from LDS (broadcast ds_load)
            const float* __restrict__ Rf = smem + i * 9;
            const float* __restrict__ tf = smem + 288 + i * 3;
            const float* __restrict__ ax = smem + 384 + i * 3;
            float F[9];
#pragma unroll
            for (int k = 0; k < 9; ++k) F[k] = Rf[k];
            const float b0 = tf[0], b1 = tf[1], b2 = tf[2];
            const float a0 = ax[0], a1 = ax[1], a2 = ax[2];

            // ---- Rodrigues about unit axis a: Rj = cI + (1-c) a a^T + s K
            float s, co;
            __sincosf(qi, &s, &co);
            const float ic = 1.0f - co;
            float J[9];
            J[0] = co + ic * a0 * a0;
            J[1] = ic * a0 * a1 - s * a2;
            J[2] = ic * a0 * a2 + s * a1;
            J[3] = ic * a0 * a1 + s * a2;
            J[4] = co + ic * a1 * a1;
            J[5] = ic * a1 * a2 - s * a0;
            J[6] = ic * a0 * a2 - s * a1;
            J[7] = ic * a1 * a2 + s * a0;
            J[8] = co + ic * a2 * a2;

            // ---- G = Rf @ Rj
            float G[9];
#pragma unroll
            for (int r = 0; r < 3; ++r)
#pragma unroll
                for (int k = 0; k < 3; ++k)
                    G[r * 3 + k] = F[r * 3 + 0] * J[0 + k]
                                 + F[r * 3 + 1] * J[3 + k]
                                 + F[r * 3 + 2] * J[6 + k];

            // ---- translation uses PARENT rotation: t = tp + Rp @ tf
            const float nt0 = tp[0] + R[0] * b0 + R[1] * b1 + R[2] * b2;
            const float nt1 = tp[1] + R[3] * b0 + R[4] * b1 + R[5] * b2;
            const float nt2 = tp[2] + R[6] * b0 + R[7] * b1 + R[8] * b2;

            // ---- Rw = Rp @ G
            float W[9];
#pragma unroll
            for (int r = 0; r < 3; ++r)
#pragma unroll
                for (int k = 0; k < 3; ++k)
                    W[r * 3 + k] = R[r * 3 + 0] * G[0 + k]
                                 + R[r * 3 + 1] * G[3 + k]
                                 + R[r * 3 + 2] * G[6 + k];

            // ---- quaternion (copysign method, matches reference)
            const float m00 = W[0], m11 = W[4], m22 = W[8];
            const float eps = 1e-12f;
            const float qw = 0.5f * sqrtf(fmaxf(1.f + m00 + m11 + m22, eps));
            float qx = 0.5f * sqrtf(fmaxf(1.f + m00 - m11 - m22, eps));
            float qy = 0.5f * sqrtf(fmaxf(1.f - m00 + m11 - m22, eps));
            float qz = 0.5f * sqrtf(fmaxf(1.f - m00 - m11 + m22, eps));
            qx *= sgnf(W[7] - W[5]);   // m21 - m12
            qy *= sgnf(W[2] - W[6]);   // m02 - m20
            qz *= sgnf(W[3] - W[1]);   // m10 - m01

            // ---- streamed, non-temporal output: out[i, b, 0..6]
            __builtin_nontemporal_store(nt0, o + 0);
            __builtin_nontemporal_store(nt1, o + 1);
            __builtin_nontemporal_store(nt2, o + 2);
            __builtin_nontemporal_store(qx,  o + 3);
            __builtin_nontemporal_store(qy,  o + 4);
            __builtin_nontemporal_store(qz,  o + 5);
            __builtin_nontemporal_store(qw,  o + 6);
            o += lstep;

            // ---- carry
            tp[0] = nt0; tp[1] = nt1; tp[2] = nt2;
#pragma unroll
            for (int k = 0; k < 9; ++k) R[k] = W[k];
        }
    }
}

extern "C" void kernel_launch(void* const* d_in, const int* in_sizes, int n_in,
                              void* d_out, int out_size, void* d_ws, size_t ws_size,
                              hipStream_t stream) {
    const float* q  = (const float*)d_in[0];  // [B, 32]
    const float* lt = (const float*)d_in[1];  // [32, 3]
    const float* lr = (const float*)d_in[2];  // [32, 3, 3]
    const float* ja = (const float*)d_in[3];  // [32, 3]
    float* out = (float*)d_out;               // [32, B, 7]

    const int nB = in_sizes[0] / NLINK;
    const int grid = (nB + TPB - 1) / TPB;
    fk_chain_kernel<<<grid, TPB, 0, stream>>>(q, lt, lr, ja, out, nB);
}